// ContrastiveLoss_66176856097301
// MI455X (gfx1250) — compile-verified
//
#include <hip/hip_runtime.h>

typedef __attribute__((ext_vector_type(2))) float v2f;
typedef __attribute__((ext_vector_type(8))) float v8f;

#define N_PTS 2048
#define D_DIM 32
#define TILE  16
#define NTILE (N_PTS / TILE)          // 128
#define NPAIRTILES (NTILE * NTILE)    // 16384

// ---------------------------------------------------------------------------
// Kernel 1: group histogram  ->  cnt[4]  (P_i = cnt[attr[i]])
// Integer shared atomics only: exact & deterministic.
// ---------------------------------------------------------------------------
__global__ void cl_hist_kernel(const int* __restrict__ attr, int* __restrict__ cnt) {
    __shared__ int scnt[4];
    const int tid = threadIdx.x;
    if (tid < 4) scnt[tid] = 0;
    __syncthreads();
    for (int i = tid; i < N_PTS; i += blockDim.x) {
        atomicAdd(&scnt[attr[i] & 3], 1);
    }
    __syncthreads();
    if (tid < 4) cnt[tid] = scnt[tid];
}

// ---------------------------------------------------------------------------
// Kernel 2: one wave32 per 16x16 pair tile.
//   - dot(p_i, p_j) over D=32 via 8x v_wmma_f32_16x16x4_f32 (full f32)
//   - max / exp-sum tail in f32 VALU (transcendental-bound part)
//   - per-tile scalar written to tileSum (no float atomics)
// ---------------------------------------------------------------------------
__global__ void __launch_bounds__(32)
cl_pair_tile_kernel(const float* __restrict__ pts, const int* __restrict__ attr,
                    const float* __restrict__ tptr, const int* __restrict__ cnt,
                    float* __restrict__ tileSum) {
    __shared__ float ldsI[TILE * D_DIM];   // 2 KB: rows i0..i0+15
    __shared__ float ldsJ[TILE * D_DIM];   // 2 KB: rows j0..j0+15
    __shared__ int   aI[TILE];
    __shared__ int   aJ[TILE];
    __shared__ float invP2[TILE];          // 1/(P_i^2) per row of the i-tile

    const int lane = threadIdx.x;          // 0..31 (wave32)
    const int i0 = blockIdx.x * TILE;
    const int j0 = blockIdx.y * TILE;
    const float invT = 1.0f / tptr[0];

    // Stage both tiles (512 floats each = 128 float4; 4 float4 per lane).
    const float4* gi = (const float4*)(pts + i0 * D_DIM);
    const float4* gj = (const float4*)(pts + j0 * D_DIM);
    float4* li4 = (float4*)ldsI;
    float4* lj4 = (float4*)ldsJ;
#pragma unroll
    for (int k = 0; k < 4; ++k) {
        li4[lane + 32 * k] = gi[lane + 32 * k];
        lj4[lane + 32 * k] = gj[lane + 32 * k];
    }
    if (lane < TILE) {
        aI[lane] = attr[i0 + lane];
        aJ[lane] = attr[j0 + lane];
    }
    __syncthreads();
    if (lane < TILE) {
        const float P = (float)cnt[aI[lane] & 3];
        invP2[lane] = 1.0f / (P * P);      // 16 divides per wave, out of hot loop
    }
    __syncthreads();

    const int half = lane >> 4;            // 0 or 1
    const int mn   = lane & 15;            // row index (A) / col index (B)
    const int kb   = half * 2;             // 32-bit A/B layout: lanes16-31 hold K+2

    // S[m][n] = dot(points[i0+m], points[j0+n]) over D=32 as 8 K=4 steps.
    // A 16x4 f32 layout (ISA 7.12.2): lane m holds {K=2*hi, K=2*hi+1}.
    // B 4x16 mirrored over N: lane n holds {B[kb][n], B[kb+1][n]}.
    // Operands read straight from LDS (lane-varying *address* -> ds_load_b64,
    // NOT a lane-varying register-array index, which would lower to cndmask
    // select chains).
    const float* aBase = &ldsI[mn * D_DIM + kb];
    const float* bBase = &ldsJ[mn * D_DIM + kb];
    v8f cacc = {};
#pragma unroll
    for (int k0 = 0; k0 < D_DIM; k0 += 4) {
        const v2f av = *(const v2f*)(aBase + k0);
        const v2f bv = *(const v2f*)(bBase + k0);
        cacc = __builtin_amdgcn_wmma_f32_16x16x4_f32(
            /*neg_a=*/false, av, /*neg_b=*/false, bv,
            /*c_mod=*/(short)0, cacc, /*reuse_a=*/false, /*reuse_b=*/false);
    }

    // Keep this lane's j-row in registers for the tail (n = mn constant
    // across r; all tail indices are static after unroll).
    float pj[D_DIM];
#pragma unroll
    for (int d = 0; d < D_DIM; ++d) pj[d] = ldsJ[mn * D_DIM + d];
    const int an = aJ[mn];

    // Softmax tail: lane owns pairs (m = r + 8*half, n = mn), matching the
    // WMMA C/D VGPR layout, so cacc[r] is consumed in place.
    float acc = 0.0f;
#pragma unroll
    for (int r = 0; r < 8; ++r) {
        const int m = r + half * 8;
        if (aI[m] == an) {
            const float* pi = &ldsI[m * D_DIM];
            float prod[D_DIM];
            float mx = -3.402823466e38f;
#pragma unroll
            for (int d = 0; d < D_DIM; ++d) {
                const float pr = pi[d] * pj[d] * invT;
                prod[d] = pr;
                mx = fmaxf(mx, pr);
            }
            float es = 0.0f;
#pragma unroll
            for (int d = 0; d < D_DIM; ++d) es += __expf(prod[d] - mx);
            const float s   = cacc[r] * invT;                    // sum_d prod
            const float rsl = s - (float)D_DIM * mx - (float)D_DIM * __logf(es);
            acc += -rsl * invP2[m];                              // 1/N applied later
        }
    }

    // Full-wave deterministic reduction -> one scalar per tile.
#pragma unroll
    for (int off = 16; off >= 1; off >>= 1) acc += __shfl_xor(acc, off, 32);
    if (lane == 0) tileSum[blockIdx.y * NTILE + blockIdx.x] = acc;
}

// ---------------------------------------------------------------------------
// Kernel 3: fixed-tree reduction of 16384 tile scalars -> final mean.
// ---------------------------------------------------------------------------
__global__ void cl_finalize_kernel(const float* __restrict__ tileSum,
                                   float* __restrict__ out) {
    __shared__ float red[256];
    const int tid = threadIdx.x;
    float s = 0.0f;
    for (int t = tid; t < NPAIRTILES; t += 256) s += tileSum[t];
    red[tid] = s;
    __syncthreads();
    for (int w = 128; w >= 1; w >>= 1) {
        if (tid < w) red[tid] += red[tid + w];
        __syncthreads();
    }
    if (tid == 0) out[0] = red[0] * (1.0f / (float)N_PTS);
}

// ---------------------------------------------------------------------------
// ws layout: [0,16)   int cnt[4]
//            [16, +16384*4) float tileSum
// ---------------------------------------------------------------------------
extern "C" void kernel_launch(void* const* d_in, const int* in_sizes, int n_in,
                              void* d_out, int out_size, void* d_ws, size_t ws_size,
                              hipStream_t stream) {
    (void)in_sizes; (void)n_in; (void)out_size; (void)ws_size;
    const float* pts  = (const float*)d_in[0];
    const int*   attr = (const int*)d_in[1];
    const float* t    = (const float*)d_in[2];
    float* out = (float*)d_out;

    int*   cnt     = (int*)d_ws;
    float* tileSum = (float*)((char*)d_ws + 16);

    cl_hist_kernel<<<1, 256, 0, stream>>>(attr, cnt);
    dim3 grid(NTILE, NTILE);
    cl_pair_tile_kernel<<<grid, 32, 0, stream>>>(pts, attr, t, cnt, tileSum);
    cl_finalize_kernel<<<1, 256, 0, stream>>>(tileSum, out);
}